// AttentionAggregator3d_35562329211816
// MI455X (gfx1250) — compile-verified
//
#include <hip/hip_runtime.h>

#define BB  2
#define CCo 128
#define CMc 128
#define PP  16
#define NN  4096   // 16*16*16

typedef float v2f __attribute__((ext_vector_type(2)));
typedef float v8f __attribute__((ext_vector_type(8)));

// D(16x16,f32) = A(16x4,f32) x B(4x16,f32) + C
__device__ __forceinline__ v8f wmma4(v2f a, v2f b, v8f c) {
    return __builtin_amdgcn_wmma_f32_16x16x4_f32(false, a, false, b, (short)0, c,
                                                 false, false);
}

// ---------------------------------------------------------------------------
// Kernel 1: transpose weights into [contraction][row] layouts so that WMMA
// fragment loads in the projection kernel are lane-contiguous.
// ---------------------------------------------------------------------------
__global__ void prep_weights(const float* __restrict__ Wq, const float* __restrict__ Wk,
                             const float* __restrict__ Wv,
                             float* __restrict__ Wqt, float* __restrict__ Wkt,
                             float* __restrict__ Wvt) {
    int i = blockIdx.x * blockDim.x + threadIdx.x;
    if (i < PP * CMc) {                      // i = c*16 + p
        int p = i & 15, c = i >> 4;
        Wqt[i] = Wq[p * CMc + c];
        Wkt[i] = Wk[p * CMc + c];
    }
    if (i < CCo * CMc) {                     // i = c*128 + o
        int o = i & 127, c = i >> 7;
        Wvt[i] = Wv[o * CMc + c];
    }
}

// --- chunk helpers: 8 contraction steps per chunk, double-buffered weights ---
__device__ __forceinline__ void chunk_loads(v2f* __restrict__ w,
                                            const float* __restrict__ Wt,
                                            int ch, int stride, int hi, int lm) {
#pragma unroll
    for (int t = 0; t < 8; ++t) {
        int c = 4 * (8 * ch + t) + 2 * hi;
        w[t].x = Wt[c * stride + lm];
        w[t].y = Wt[(c + 1) * stride + lm];
    }
}
__device__ __forceinline__ void chunk_wmma_qk(const v2f* __restrict__ w,
                                              const v2f* __restrict__ Z, int ch,
                                              v8f& d0, v8f& d1) {
#pragma unroll
    for (int t = 0; t < 8; t += 2) {
        d0 = wmma4(w[t],     Z[8 * ch + t],     d0);
        d1 = wmma4(w[t + 1], Z[8 * ch + t + 1], d1);
    }
}
__device__ __forceinline__ void chunk_wmma_v(const v2f* __restrict__ w,
                                             const v2f* __restrict__ Z, int ch,
                                             v8f& d0, v8f& d1) {
#pragma unroll
    for (int t = 0; t < 8; t += 2) {
        d0 = wmma4(Z[8 * ch + t],     w[t],     d0);
        d1 = wmma4(Z[8 * ch + t + 1], w[t + 1], d1);
    }
}

// ---------------------------------------------------------------------------
// Kernel 2: Q/K/V projections. One wave per (batch, 16-token tile).
// zm tile kept resident (64 VGPRs); weights streamed in double-buffered
// 8-step chunks; sched barriers bound live ranges so nothing spills.
// ---------------------------------------------------------------------------
__global__ void __launch_bounds__(256, 1)
proj_qkv(const float* __restrict__ zm,
         const float* __restrict__ Wqt, const float* __restrict__ Wkt,
         const float* __restrict__ Wvt,
         const float* __restrict__ bq, const float* __restrict__ bk,
         const float* __restrict__ bv,
         float* __restrict__ Q, float* __restrict__ K, float* __restrict__ Vt) {
    const int lane = threadIdx.x & 31;
    const int lm   = lane & 15;
    const int hi   = lane >> 4;                       // lane half (K-split)
    const int gw   = blockIdx.x * 8 + (threadIdx.x >> 5);
    const int b    = gw >> 8;                         // 256 tiles per batch
    const int n0   = (gw & 255) << 4;

    const float* zmb = zm + (size_t)b * CMc * NN;

    v2f Z[32];
#pragma unroll
    for (int kk = 0; kk < 32; ++kk) {
        int c = 4 * kk + 2 * hi;
        Z[kk].x = zmb[(size_t)c       * NN + n0 + lm];
        Z[kk].y = zmb[(size_t)(c + 1) * NN + n0 + lm];
    }
    __builtin_amdgcn_sched_barrier(0);

    v2f w0[8], w1[8];
    // ---- Q = Wq x zm : D[p][n] ----
    {
        v8f d0 = {}, d1 = {};
        chunk_loads(w0, Wqt, 0, PP, hi, lm);
        __builtin_amdgcn_sched_barrier(0);
        chunk_loads(w1, Wqt, 1, PP, hi, lm);
        chunk_wmma_qk(w0, Z, 0, d0, d1);
        __builtin_amdgcn_sched_barrier(0);
        chunk_loads(w0, Wqt, 2, PP, hi, lm);
        chunk_wmma_qk(w1, Z, 1, d0, d1);
        __builtin_amdgcn_sched_barrier(0);
        chunk_loads(w1, Wqt, 3, PP, hi, lm);
        chunk_wmma_qk(w0, Z, 2, d0, d1);
        __builtin_amdgcn_sched_barrier(0);
        chunk_wmma_qk(w1, Z, 3, d0, d1);
        __builtin_amdgcn_sched_barrier(0);
        float* Qb = Q + (size_t)b * PP * NN;
#pragma unroll
        for (int r = 0; r < 8; ++r) {
            int p = r + 8 * hi;
            Qb[(size_t)p * NN + n0 + lm] = d0[r] + d1[r] + bq[p];
        }
        __builtin_amdgcn_sched_barrier(0);
    }
    // ---- K = Wk x zm : D[p][n] ----
    {
        v8f d0 = {}, d1 = {};
        chunk_loads(w0, Wkt, 0, PP, hi, lm);
        __builtin_amdgcn_sched_barrier(0);
        chunk_loads(w1, Wkt, 1, PP, hi, lm);
        chunk_wmma_qk(w0, Z, 0, d0, d1);
        __builtin_amdgcn_sched_barrier(0);
        chunk_loads(w0, Wkt, 2, PP, hi, lm);
        chunk_wmma_qk(w1, Z, 1, d0, d1);
        __builtin_amdgcn_sched_barrier(0);
        chunk_loads(w1, Wkt, 3, PP, hi, lm);
        chunk_wmma_qk(w0, Z, 2, d0, d1);
        __builtin_amdgcn_sched_barrier(0);
        chunk_wmma_qk(w1, Z, 3, d0, d1);
        __builtin_amdgcn_sched_barrier(0);
        float* Kb = K + (size_t)b * PP * NN;
#pragma unroll
        for (int r = 0; r < 8; ++r) {
            int p = r + 8 * hi;
            Kb[(size_t)p * NN + n0 + lm] = d0[r] + d1[r] + bk[p];
        }
        __builtin_amdgcn_sched_barrier(0);
    }
    // ---- Vt = zm^T x Wv^T : D[n][o] ----
    {
        float* Vtb = Vt + (size_t)b * NN * CCo;
#pragma unroll 1
        for (int ot = 0; ot < 8; ++ot) {
            const float* Wvo = Wvt + ot * 16;         // column block of Wv^T
            v8f d0 = {}, d1 = {};
            chunk_loads(w0, Wvo, 0, CCo, hi, lm);
            __builtin_amdgcn_sched_barrier(0);
            chunk_loads(w1, Wvo, 1, CCo, hi, lm);
            chunk_wmma_v(w0, Z, 0, d0, d1);
            __builtin_amdgcn_sched_barrier(0);
            chunk_loads(w0, Wvo, 2, CCo, hi, lm);
            chunk_wmma_v(w1, Z, 1, d0, d1);
            __builtin_amdgcn_sched_barrier(0);
            chunk_loads(w1, Wvo, 3, CCo, hi, lm);
            chunk_wmma_v(w0, Z, 2, d0, d1);
            __builtin_amdgcn_sched_barrier(0);
            chunk_wmma_v(w1, Z, 3, d0, d1);
            __builtin_amdgcn_sched_barrier(0);
            float bvv = bv[ot * 16 + lm];
#pragma unroll
            for (int r = 0; r < 8; ++r) {
                int n = n0 + r + 8 * hi;
                Vtb[(size_t)n * CCo + ot * 16 + lm] = d0[r] + d1[r] + bvv;
            }
            __builtin_amdgcn_sched_barrier(0);
        }
    }
}

// ---------------------------------------------------------------------------
// Kernel 3: flash attention + residual. One wave per (batch, 16-query tile).
//   - K fragments double-buffered (ka/kb); next K issued before this tile's
//     V loads (loads complete in order => K(j+1) resident when PV waits).
//   - all 64 V-fragment loads issued before the softmax so their latency is
//     covered by max/exp/rescale/LDS-transpose work.
//   - sched barrier at step end: step N+1's loads can't hoist over PV(N).
// ---------------------------------------------------------------------------
__device__ __forceinline__ void attn_step(
    int n0, int npre,
    const float* __restrict__ Kb, const float* __restrict__ Vtb,
    v2f* __restrict__ ka, v2f* __restrict__ kb, const v2f* __restrict__ qm,
    v8f* __restrict__ acc, float& mrow, float& lrow,
    float* __restrict__ pt, int lm, int hi) {

    // logits tile St[n][m] from pre-loaded K fragments (2 interleaved chains)
    v8f st0 = {}, st1 = {};
    st0 = wmma4(ka[0], qm[0], st0);
    st1 = wmma4(ka[1], qm[1], st1);
    st0 = wmma4(ka[2], qm[2], st0);
    st1 = wmma4(ka[3], qm[3], st1);

    // prefetch next K tile (issued before V so it completes first: in-order)
#pragma unroll
    for (int kk = 0; kk < 4; ++kk) {
        int p = 4 * kk + 2 * hi;
        kb[kk].x = Kb[(size_t)p       * NN + npre + lm];
        kb[kk].y = Kb[(size_t)(p + 1) * NN + npre + lm];
    }

    // V tile fragments: issue all loads now, consume after softmax
    v2f va[8][4];
#pragma unroll
    for (int ot = 0; ot < 8; ++ot)
#pragma unroll
        for (int kk = 0; kk < 4; ++kk) {
            int nr = n0 + 4 * kk + 2 * hi;
            va[ot][kk].x = Vtb[(size_t)nr       * CCo + ot * 16 + lm];
            va[ot][kk].y = Vtb[(size_t)(nr + 1) * CCo + ot * 16 + lm];
        }

    v8f st;
#pragma unroll
    for (int r = 0; r < 8; ++r) st[r] = st0[r] + st1[r];

    // online softmax over n (8 VGPR rows + one cross-half shuffle)
    float mx = st[0];
#pragma unroll
    for (int r = 1; r < 8; ++r) mx = fmaxf(mx, st[r]);
    mx = fmaxf(mx, __shfl_xor(mx, 16, 32));
    float mnew  = fmaxf(mrow, mx);
    float alpha = __expf(mrow - mnew);
    float s = 0.0f;
#pragma unroll
    for (int r = 0; r < 8; ++r) {
        float e = __expf(st[r] - mnew);
        st[r] = e;
        s += e;
    }
    s += __shfl_xor(s, 16, 32);
    lrow = lrow * alpha + s;
    mrow = mnew;
#pragma unroll
    for (int ot = 0; ot < 8; ++ot)
#pragma unroll
        for (int r = 0; r < 8; ++r) acc[ot][r] *= alpha;

    // P tile -> LDS [m][n] (stride 18 floats: conflict-free, 8B-aligned pairs);
    // same-wave DS ops are in-order, tile is wave-private: no barrier needed.
#pragma unroll
    for (int r = 0; r < 8; ++r) pt[lm * 18 + r + 8 * hi] = st[r];
    v2f bb[4];
#pragma unroll
    for (int kk = 0; kk < 4; ++kk) {
        bb[kk].x = pt[lm * 18 + 4 * kk + 2 * hi];
        bb[kk].y = pt[lm * 18 + 4 * kk + 2 * hi + 1];
    }

    // O[o][m] += V[o][n] * P[m][n] : 8 independent 4-step WMMA chains
#pragma unroll
    for (int ot = 0; ot < 8; ++ot)
#pragma unroll
        for (int kk = 0; kk < 4; ++kk)
            acc[ot] = wmma4(va[ot][kk], bb[kk], acc[ot]);

    __builtin_amdgcn_sched_barrier(0);   // keep next step's loads below PV
}

__global__ void __launch_bounds__(256, 1)
attn_out(const float* __restrict__ Q, const float* __restrict__ K,
         const float* __restrict__ Vt, const float* __restrict__ zc,
         const float* __restrict__ gamma, float* __restrict__ out) {
    __shared__ float lds[8 * 16 * 18];                // per-wave 16x16 tile, stride 18
    const int lane = threadIdx.x & 31;
    const int lm   = lane & 15;
    const int hi   = lane >> 4;
    const int wib  = threadIdx.x >> 5;
    const int gw   = blockIdx.x * 8 + wib;
    const int b    = gw >> 8;
    const int m0   = (gw & 255) << 4;
    float* pt = lds + wib * (16 * 18);

    const float* Qb  = Q  + (size_t)b * PP * NN;
    const float* Kb  = K  + (size_t)b * PP * NN;
    const float* Vtb = Vt + (size_t)b * NN * CCo;

    // Q fragment for this m-tile: B[p][m], reused across all key tiles
    v2f qm[4];
#pragma unroll
    for (int kk = 0; kk < 4; ++kk) {
        int p = 4 * kk + 2 * hi;
        qm[kk].x = Qb[(size_t)p       * NN + m0 + lm];
        qm[kk].y = Qb[(size_t)(p + 1) * NN + m0 + lm];
    }

    v8f acc[8];
#pragma unroll
    for (int ot = 0; ot < 8; ++ot) acc[ot] = (v8f){};
    float mrow = -1.0e30f;
    float lrow = 0.0f;

    // preload K tile 0
    v2f ka[4], kb[4];
#pragma unroll
    for (int kk = 0; kk < 4; ++kk) {
        int p = 4 * kk + 2 * hi;
        ka[kk].x = Kb[(size_t)p       * NN + lm];
        ka[kk].y = Kb[(size_t)(p + 1) * NN + lm];
    }

    for (int j = 0; j < NN / 16; j += 2) {
        int np1 = (j + 1) * 16;                        // j+1 <= 255 here
        int np2 = (j + 2 < NN / 16 ? j + 2 : NN / 16 - 1) * 16;
        attn_step(j * 16, np1, Kb, Vtb, ka, kb, qm, acc, mrow, lrow, pt, lm, hi);
        attn_step(np1,    np2, Kb, Vtb, kb, ka, qm, acc, mrow, lrow, pt, lm, hi);
    }

    const float inv = 1.0f / lrow;
    const float g   = gamma[0];
#pragma unroll
    for (int ot = 0; ot < 8; ++ot) {
#pragma unroll
        for (int r = 0; r < 8; ++r) {
            int o = ot * 16 + r + 8 * hi;
            size_t idx = ((size_t)b * CCo + o) * NN + m0 + lm;
            out[idx] = zc[idx] + g * acc[ot][r] * inv;
        }
    }
}

// ---------------------------------------------------------------------------
extern "C" void kernel_launch(void* const* d_in, const int* in_sizes, int n_in,
                              void* d_out, int out_size, void* d_ws, size_t ws_size,
                              hipStream_t stream) {
    const float* zc    = (const float*)d_in[0];
    const float* zm    = (const float*)d_in[1];
    const float* Wq    = (const float*)d_in[2];
    const float* bq    = (const float*)d_in[3];
    const float* Wk    = (const float*)d_in[4];
    const float* bk    = (const float*)d_in[5];
    const float* Wv    = (const float*)d_in[6];
    const float* bv    = (const float*)d_in[7];
    const float* gamma = (const float*)d_in[8];
    float* out = (float*)d_out;

    float* ws  = (float*)d_ws;
    float* Wqt = ws;                                  //   128*16
    float* Wkt = Wqt + PP * CMc;                      //   128*16
    float* Wvt = Wkt + PP * CMc;                      //   128*128
    float* Qs  = Wvt + CCo * CMc;                     //   2*16*4096
    float* Ks  = Qs + (size_t)BB * PP * NN;           //   2*16*4096
    float* Vts = Ks + (size_t)BB * PP * NN;           //   2*4096*128  (~5.3 MB total)

    prep_weights<<<64, 256, 0, stream>>>(Wq, Wk, Wv, Wqt, Wkt, Wvt);
    proj_qkv<<<64, 256, 0, stream>>>(zm, Wqt, Wkt, Wvt, bq, bk, bv, Qs, Ks, Vts);
    attn_out<<<64, 256, 0, stream>>>(Qs, Ks, Vts, zc, gamma, out);
}